// GNBlock_76914274337220
// MI455X (gfx1250) — compile-verified
//
#include <hip/hip_runtime.h>

#define N_NODES 100000
#define N_EDGES 1600000
#define D_IN 64
#define D_OUT 64
#define K_DIM 128
#define BN_EPS 1e-5f
#define WT_STRIDE 132  // 128 + 4 pad: conflict-free transposed ds_load_b64

typedef __attribute__((ext_vector_type(2))) float v2f;
typedef __attribute__((ext_vector_type(8))) float v8f;

// fp32 WMMA: D(16x16,f32) = A(16x4,f32) x B(4x16,f32) + C
// A layout: lane l -> M = l%16, VGPR j holds K = 2*(l/16)+j
// B layout (mirror): lane l -> N = l%16, VGPR j holds K = 2*(l/16)+j
// C/D layout: VGPR r, lanes 0-15: M=r, N=lane; lanes 16-31: M=r+8, N=lane-16
static __device__ __forceinline__ v8f wmma4(v2f a, v2f b, v8f c) {
  return __builtin_amdgcn_wmma_f32_16x16x4_f32(false, a, false, b, (short)0, c,
                                               false, false);
}

static __device__ __forceinline__ void atomAddG(float* p, float v) {
  __hip_atomic_fetch_add(p, v, __ATOMIC_RELAXED, __HIP_MEMORY_SCOPE_AGENT);
}
static __device__ __forceinline__ void atomAddW(float* p, float v) {
  __hip_atomic_fetch_add(p, v, __ATOMIC_RELAXED, __HIP_MEMORY_SCOPE_WORKGROUP);
}

// Load row-major W[K_DIM x D_OUT] from global, store transposed+padded into
// LDS as Wt[n * WT_STRIDE + k]. B fragments then become contiguous b64 reads.
static __device__ __forceinline__ void load_w_transposed(
    float* Wt, const float* __restrict__ W, int tid) {
  for (int i = tid; i < (K_DIM * D_OUT) / 4; i += 256) {
    const int k = (i * 4) >> 6;  // row of W
    const int n = (i * 4) & 63;  // col of W
    const float4 w = *(const float4*)&W[i * 4];
    Wt[(n + 0) * WT_STRIDE + k] = w.x;
    Wt[(n + 1) * WT_STRIDE + k] = w.y;
    Wt[(n + 2) * WT_STRIDE + k] = w.z;
    Wt[(n + 3) * WT_STRIDE + k] = w.w;
  }
}

__global__ void gn_zero_kernel(float* __restrict__ p, int n) {
  int i = blockIdx.x * blockDim.x + threadIdx.x;
  if (i < n) p[i] = 0.0f;
}

// ---------------------------------------------------------------------------
// Edge kernel: msg = concat(x[src], x[dst]) @ W_edge + b_edge, atomically
// scatter-summed into agg[dst]. 16 edges per wave-tile, 4 tiles per wave.
// ---------------------------------------------------------------------------
__global__ __launch_bounds__(256) void gn_edge_kernel(
    const float* __restrict__ x, const long long* __restrict__ ei,
    const float* __restrict__ We, const float* __restrict__ be,
    float* __restrict__ agg, int nTiles) {
  __shared__ float Wt[D_OUT * WT_STRIDE];  // ~33 KB, transposed + padded
  __shared__ float bl[D_OUT];
  const int tid = threadIdx.x;
  load_w_transposed(Wt, We, tid);
  if (tid < D_OUT) bl[tid] = be[tid];
  __syncthreads();

  const int lane = tid & 31;
  const int wave = tid >> 5;
  const int m = lane & 15;     // matrix row/col index within 16
  const int half = lane >> 4;  // selects K pair
  const int tile0 = (blockIdx.x * 8 + wave) * 4;
  const float* wbase = &Wt[m * WT_STRIDE];

  for (int t = 0; t < 4; ++t) {
    const int tile = tile0 + t;  // wave-uniform guard -> EXEC all ones at WMMA
    if (tile >= nTiles) break;
    const int e = tile * 16 + m;
    const long long s = ei[e];
    const int di = (int)ei[N_EDGES + e];
    const float* xs = x + s * (long long)D_IN;
    const float* xd = x + (long long)di * D_IN;

    v8f acc0 = {}, acc1 = {}, acc2 = {}, acc3 = {};
#pragma unroll
    for (int kk = 0; kk < 32; ++kk) {
      const int c0 = kk * 4 + half * 2;
      // first 64 K-columns come from x[src], last 64 from x[dst]
      const float* arow = (kk < 16) ? (xs + c0) : (xd + (c0 - 64));
      const float2 af = *(const float2*)arow;
      v2f a;
      a.x = af.x;
      a.y = af.y;
      const float* wp = wbase + c0;
      const v2f b0 = *(const v2f*)(wp + 0 * 16 * WT_STRIDE);
      acc0 = wmma4(a, b0, acc0);
      const v2f b1 = *(const v2f*)(wp + 1 * 16 * WT_STRIDE);
      acc1 = wmma4(a, b1, acc1);
      const v2f b2 = *(const v2f*)(wp + 2 * 16 * WT_STRIDE);
      acc2 = wmma4(a, b2, acc2);
      const v2f b3 = *(const v2f*)(wp + 3 * 16 * WT_STRIDE);
      acc3 = wmma4(a, b3, acc3);
    }

    const float bb0 = bl[m], bb1 = bl[m + 16], bb2 = bl[m + 32],
                bb3 = bl[m + 48];
#pragma unroll
    for (int r = 0; r < 8; ++r) {
      // C row M = r + 8*half is held by lanes whose m == M -> shuffle dst idx
      const int dr = __shfl(di, r + 8 * half, 32);
      float* ap = agg + (long long)dr * D_OUT + m;
      atomAddG(ap + 0, acc0[r] + bb0);
      atomAddG(ap + 16, acc1[r] + bb1);
      atomAddG(ap + 32, acc2[r] + bb2);
      atomAddG(ap + 48, acc3[r] + bb3);
    }
  }
}

// ---------------------------------------------------------------------------
// Node kernel: h = PReLU(concat(x, agg) @ W_node + b_node); write h to hout,
// accumulate per-channel sum(h), sum(h^2) for batch-norm.
// ---------------------------------------------------------------------------
__global__ __launch_bounds__(256) void gn_node_kernel(
    const float* __restrict__ x, const float* __restrict__ agg,
    const float* __restrict__ Wn, const float* __restrict__ bn,
    const float* __restrict__ prelu_a, float* __restrict__ hout,
    float* __restrict__ sums, int nTiles) {
  __shared__ float Wt[D_OUT * WT_STRIDE];  // ~33 KB, transposed + padded
  __shared__ float bl[D_OUT];
  __shared__ float sS[D_OUT];
  __shared__ float sQ[D_OUT];
  const int tid = threadIdx.x;
  load_w_transposed(Wt, Wn, tid);
  if (tid < D_OUT) {
    bl[tid] = bn[tid];
    sS[tid] = 0.0f;
    sQ[tid] = 0.0f;
  }
  __syncthreads();
  const float slope = *prelu_a;

  const int lane = tid & 31;
  const int wave = tid >> 5;
  const int m = lane & 15;
  const int half = lane >> 4;
  const int tile0 = (blockIdx.x * 8 + wave) * 4;
  const float* wbase = &Wt[m * WT_STRIDE];

  for (int t = 0; t < 4; ++t) {
    const int tile = tile0 + t;  // wave-uniform
    if (tile >= nTiles) break;
    const int node = tile * 16 + m;
    const float* xr = x + (long long)node * D_IN;
    const float* ar = agg + (long long)node * D_OUT;

    v8f acc0 = {}, acc1 = {}, acc2 = {}, acc3 = {};
#pragma unroll
    for (int kk = 0; kk < 32; ++kk) {
      const int c0 = kk * 4 + half * 2;
      const float* arow = (kk < 16) ? (xr + c0) : (ar + (c0 - 64));
      const float2 af = *(const float2*)arow;
      v2f a;
      a.x = af.x;
      a.y = af.y;
      const float* wp = wbase + c0;
      const v2f b0 = *(const v2f*)(wp + 0 * 16 * WT_STRIDE);
      acc0 = wmma4(a, b0, acc0);
      const v2f b1 = *(const v2f*)(wp + 1 * 16 * WT_STRIDE);
      acc1 = wmma4(a, b1, acc1);
      const v2f b2 = *(const v2f*)(wp + 2 * 16 * WT_STRIDE);
      acc2 = wmma4(a, b2, acc2);
      const v2f b3 = *(const v2f*)(wp + 3 * 16 * WT_STRIDE);
      acc3 = wmma4(a, b3, acc3);
    }

    const float bb0 = bl[m], bb1 = bl[m + 16], bb2 = bl[m + 32],
                bb3 = bl[m + 48];
    float p0 = 0.f, p1 = 0.f, p2 = 0.f, p3 = 0.f;
    float q0 = 0.f, q1 = 0.f, q2 = 0.f, q3 = 0.f;
#pragma unroll
    for (int r = 0; r < 8; ++r) {
      const int row = tile * 16 + r + 8 * half;
      float* op = hout + (long long)row * D_OUT + m;
      float h0 = acc0[r] + bb0; h0 = h0 >= 0.f ? h0 : slope * h0;
      op[0] = h0;  p0 += h0; q0 += h0 * h0;
      float h1 = acc1[r] + bb1; h1 = h1 >= 0.f ? h1 : slope * h1;
      op[16] = h1; p1 += h1; q1 += h1 * h1;
      float h2 = acc2[r] + bb2; h2 = h2 >= 0.f ? h2 : slope * h2;
      op[32] = h2; p2 += h2; q2 += h2 * h2;
      float h3 = acc3[r] + bb3; h3 = h3 >= 0.f ? h3 : slope * h3;
      op[48] = h3; p3 += h3; q3 += h3 * h3;
    }
    atomAddW(&sS[m], p0);      atomAddW(&sQ[m], q0);
    atomAddW(&sS[m + 16], p1); atomAddW(&sQ[m + 16], q1);
    atomAddW(&sS[m + 32], p2); atomAddW(&sQ[m + 32], q2);
    atomAddW(&sS[m + 48], p3); atomAddW(&sQ[m + 48], q3);
  }
  __syncthreads();
  if (tid < D_OUT) {
    atomAddG(&sums[tid], sS[tid]);
    atomAddG(&sums[D_OUT + tid], sQ[tid]);
  }
}

// ---------------------------------------------------------------------------
// BatchNorm finalize (in-place on hout): (h-mean)*rsqrt(var+eps)*gamma+beta
// ---------------------------------------------------------------------------
__global__ void gn_bn_kernel(float* __restrict__ h,
                             const float* __restrict__ sums,
                             const float* __restrict__ gamma,
                             const float* __restrict__ beta, int total) {
  const int idx = blockIdx.x * blockDim.x + threadIdx.x;
  if (idx >= total) return;
  const int c = idx & (D_OUT - 1);
  const float inv_n = 1.0f / (float)N_NODES;
  const float mean = sums[c] * inv_n;
  const float var = sums[D_OUT + c] * inv_n - mean * mean;
  const float inv = rsqrtf(var + BN_EPS);
  const float v = h[idx];
  h[idx] = (v - mean) * inv * gamma[c] + beta[c];
}

extern "C" void kernel_launch(void* const* d_in, const int* in_sizes, int n_in,
                              void* d_out, int out_size, void* d_ws,
                              size_t ws_size, hipStream_t stream) {
  const float* x = (const float*)d_in[0];
  const long long* ei = (const long long*)d_in[1];
  const float* We = (const float*)d_in[2];
  const float* be = (const float*)d_in[3];
  const float* Wn = (const float*)d_in[4];
  const float* bnb = (const float*)d_in[5];
  const float* pa = (const float*)d_in[6];
  const float* gamma = (const float*)d_in[7];
  const float* beta = (const float*)d_in[8];
  float* out = (float*)d_out;

  float* agg = (float*)d_ws;                    // N_NODES * 64 floats
  float* sums = agg + (size_t)N_NODES * D_OUT;  // 128 floats (sum, sumsq)

  const int zn = N_NODES * D_OUT + 2 * D_OUT;
  hipLaunchKernelGGL(gn_zero_kernel, dim3((zn + 255) / 256), dim3(256), 0,
                     stream, agg, zn);

  const int eTiles = N_EDGES / 16;  // 100000
  hipLaunchKernelGGL(gn_edge_kernel, dim3((eTiles + 31) / 32), dim3(256), 0,
                     stream, x, ei, We, be, agg, eTiles);

  const int nTiles = N_NODES / 16;  // 6250
  hipLaunchKernelGGL(gn_node_kernel, dim3((nTiles + 31) / 32), dim3(256), 0,
                     stream, x, agg, Wn, bnb, pa, out, sums, nTiles);

  const int total = N_NODES * D_OUT;
  hipLaunchKernelGGL(gn_bn_kernel, dim3((total + 255) / 256), dim3(256), 0,
                     stream, out, sums, gamma, beta, total);
}